// NST_KDLoss_48258252538614
// MI455X (gfx1250) — compile-verified
//
#include <hip/hip_runtime.h>
#include <hip/hip_bf16.h>

typedef __attribute__((ext_vector_type(2))) float v2f;
typedef __attribute__((ext_vector_type(4))) float v4f;
typedef __attribute__((ext_vector_type(8))) float v8f;

#define B_DIM 32
#define S_DIM 8192
#define D_DIM 128
#define NCHUNK 8
#define CH_ROWS (S_DIM / NCHUNK)   // 1024 rows per workgroup
#define BLK_ROWS 64                // rows staged in LDS per iteration
#define NBLK (CH_ROWS / BLK_ROWS)  // 16
#define LDS_STRIDE 66              // padded column stride (floats): b64-aligned, bank-spread

// ---------------- zero init (d_out and workspace are poisoned) ----------------
__global__ void nst_zero(float* wsG, float* out, int n) {
    int i = blockIdx.x * blockDim.x + threadIdx.x;
    if (i < n) wsG[i] = 0.0f;
    if (i == 0) out[0] = 0.0f;
}

// ---------------- phase 1: partial G = X^T X via f32 WMMA + column sums ----------------
__global__ __launch_bounds__(256) void nst_phase1(const float* __restrict__ X,
                                                  float* __restrict__ wsG,
                                                  float* __restrict__ wsXsum) {
    __shared__ float lds[D_DIM * LDS_STRIDE];   // 128*66*4 = 33,792 bytes (column-major, padded)

    const int b     = blockIdx.y;
    const int chunk = blockIdx.x;
    const int tid   = threadIdx.x;
    const int w     = tid >> 5;     // wave id 0..7 (wave32)
    const int lane  = tid & 31;

    const float* Xb = X + (size_t)b * S_DIM * D_DIM + (size_t)chunk * CH_ROWS * D_DIM;

    // 8 accumulator tiles (mt = w fixed, nt = 0..7): 64 VGPRs
    v8f acc[8];
#pragma unroll
    for (int i = 0; i < 8; ++i) acc[i] = (v8f){0.f,0.f,0.f,0.f,0.f,0.f,0.f,0.f};

    // staging geometry: thread covers float4-column f4c, rows {row0, row0+8, ...}
    const int f4c  = tid & 31;      // 0..31  -> columns 4*f4c .. 4*f4c+3
    const int row0 = tid >> 5;      // 0..7
    v4f colsum = (v4f){0.f, 0.f, 0.f, 0.f};

    const int mhalf = lane >> 4;    // 0/1 -> K pair {0,1} or {2,3}
    const int ln    = lane & 15;    // M/N within tile

    // -------- software pipeline: prefetch block 0 into registers --------
    v4f pref[8];
#pragma unroll
    for (int p = 0; p < 8; ++p) {
        const int r = row0 + p * 8;
        pref[p] = *(const v4f*)(Xb + (size_t)r * D_DIM + f4c * 4);
    }

    for (int blk = 0; blk < NBLK; ++blk) {
        __syncthreads();            // previous block's LDS readers are done
        // ---- commit prefetched 64x128 block, transposed, into LDS; fold column sums ----
#pragma unroll
        for (int p = 0; p < 8; ++p) {
            const int r  = row0 + p * 8;
            const v4f val = pref[p];
            colsum += val;
            const int c0 = f4c * 4;
            lds[(c0 + 0) * LDS_STRIDE + r] = val.x;
            lds[(c0 + 1) * LDS_STRIDE + r] = val.y;
            lds[(c0 + 2) * LDS_STRIDE + r] = val.z;
            lds[(c0 + 3) * LDS_STRIDE + r] = val.w;
        }
        __syncthreads();
        // ---- issue next block's global loads (overlap with WMMA compute below) ----
        if (blk + 1 < NBLK) {
#pragma unroll
            for (int p = 0; p < 8; ++p) {
                const int r = row0 + p * 8;
                pref[p] = *(const v4f*)(Xb + (size_t)((blk + 1) * BLK_ROWS + r) * D_DIM + f4c * 4);
            }
        }
        // ---- compute: 16 K-steps of 4, 8 WMMA tiles per step ----
#pragma unroll
        for (int kk = 0; kk < BLK_ROWS / 4; ++kk) {
            const int s0k = kk * 4 + mhalf * 2;    // even -> 8B-aligned ds_load_b64
            v2f bmat[8];
#pragma unroll
            for (int nt = 0; nt < 8; ++nt)
                bmat[nt] = *(const v2f*)&lds[(nt * 16 + ln) * LDS_STRIDE + s0k];
            v2f amat = *(const v2f*)&lds[(w * 16 + ln) * LDS_STRIDE + s0k];
#pragma unroll
            for (int nt = 0; nt < 8; ++nt) {
                acc[nt] = __builtin_amdgcn_wmma_f32_16x16x4_f32(
                    /*neg_a=*/false, amat, /*neg_b=*/false, bmat[nt],
                    /*c_mod=*/(short)0, acc[nt], /*reuse_a=*/false, /*reuse_b=*/false);
            }
        }
    }

    // ---- accumulate partial gram into workspace ----
    float* Gb = wsG + (size_t)b * D_DIM * D_DIM;
#pragma unroll
    for (int nt = 0; nt < 8; ++nt) {
#pragma unroll
        for (int v = 0; v < 8; ++v) {
            const int mrow = w * 16 + mhalf * 8 + v;   // C layout: VGPR v -> M=v (+8 for upper half-wave)
            const int ncol = nt * 16 + ln;
            atomicAdd(&Gb[mrow * D_DIM + ncol], acc[nt][v]);
        }
    }
    // ---- accumulate column sums ----
    float* xsb = wsXsum + b * D_DIM;
    const int c0 = f4c * 4;
    atomicAdd(&xsb[c0 + 0], colsum.x);
    atomicAdd(&xsb[c0 + 1], colsum.y);
    atomicAdd(&xsb[c0 + 2], colsum.z);
    atomicAdd(&xsb[c0 + 3], colsum.w);
}

// ---------------- phase 2: Frobenius + closed-form teacher terms ----------------
__global__ __launch_bounds__(256) void nst_phase2(const float* __restrict__ wsG,
                                                  const float* __restrict__ wsXsum,
                                                  const float* __restrict__ teacher,
                                                  float* __restrict__ out) {
    __shared__ float red[256];
    const int b = blockIdx.x;
    const int tid = threadIdx.x;

    const float* Gb = wsG + (size_t)b * D_DIM * D_DIM;
    float fro = 0.f;
    for (int i = tid; i < D_DIM * D_DIM; i += 256) { float g = Gb[i]; fro += g * g; }

    float x2 = 0.f, t2 = 0.f;
    if (tid < D_DIM) {
        float xs = wsXsum[b * D_DIM + tid];  x2 = xs * xs;
        float tv = teacher[b * D_DIM + tid]; t2 = tv * tv;
    }

    // reduction 1: Frobenius
    red[tid] = fro; __syncthreads();
    for (int s = 128; s > 0; s >>= 1) { if (tid < s) red[tid] += red[tid + s]; __syncthreads(); }
    const float froT = red[0]; __syncthreads();
    // reduction 2: ||x_sum||^2
    red[tid] = x2; __syncthreads();
    for (int s = 128; s > 0; s >>= 1) { if (tid < s) red[tid] += red[tid + s]; __syncthreads(); }
    const float x2T = red[0]; __syncthreads();
    // reduction 3: ||t||^2
    red[tid] = t2; __syncthreads();
    for (int s = 128; s > 0; s >>= 1) { if (tid < s) red[tid] += red[tid + s]; __syncthreads(); }
    const float t2T = red[0];

    if (tid == 0) {
        const float Sf = (float)S_DIM;
        const float per = froT + Sf * Sf * t2T * t2T - 2.0f * x2T * t2T;
        const float scale = 1.0f / ((float)B_DIM * (float)D_DIM * (float)D_DIM);
        atomicAdd(out, per * scale);
    }
}

extern "C" void kernel_launch(void* const* d_in, const int* in_sizes, int n_in,
                              void* d_out, int out_size, void* d_ws, size_t ws_size,
                              hipStream_t stream) {
    const float* student = (const float*)d_in[0];   // [32, 8192, 128] f32
    const float* teacher = (const float*)d_in[1];   // [32, 128] f32
    float* out = (float*)d_out;                     // scalar f32

    float* wsG    = (float*)d_ws;                                   // 32*128*128 floats
    float* wsXsum = wsG + (size_t)B_DIM * D_DIM * D_DIM;            // 32*128 floats
    const int nzero = B_DIM * D_DIM * D_DIM + B_DIM * D_DIM;        // 528,384 floats (~2.1 MB)

    nst_zero<<<(nzero + 255) / 256, 256, 0, stream>>>(wsG, out, nzero);
    nst_phase1<<<dim3(NCHUNK, B_DIM), 256, 0, stream>>>(student, wsG, wsXsum);
    nst_phase2<<<B_DIM, 256, 0, stream>>>(wsG, wsXsum, teacher, out);
}